// HK_layer_shuffle_5025111736794
// MI455X (gfx1250) — compile-verified
//
#include <hip/hip_runtime.h>
#include <hip/hip_bf16.h>

typedef float v2f __attribute__((ext_vector_type(2)));
typedef float v8f __attribute__((ext_vector_type(8)));

#define NN   8192
#define C2   128
#define MM   16          /* M*M */
#define QR   4096        /* B * M * M rows */
#define TK   40
#define CH   128         /* k-chunk for applied pipeline */
#define NCHUNK (NN / CH)
#define EPSF 1e-5f

__device__ __forceinline__ v8f wmma4(v2f a, v2f b, v8f c) {
  // V_WMMA_F32_16X16X4_F32: D = A(16x4) * B(4x16) + C(16x16), fp32 throughout
  return __builtin_amdgcn_wmma_f32_16x16x4_f32(false, a, false, b, (short)0, c, false, false);
}

// CDNA5 async global->LDS copy (ASYNCcnt-tracked, see CDNA5 ISA)
__device__ __forceinline__ void async_b128(unsigned int lds_off, const float* g) {
  asm volatile("global_load_async_to_lds_b128 %0, %1, off"
               :: "v"(lds_off), "v"(g) : "memory");
}

// ---------------- q = x(1024x128) @ Wq^T(128x512) + bq ----------------
__global__ void k_qraw(const float* __restrict__ x, const float* __restrict__ Wq,
                       const float* __restrict__ bq, float* __restrict__ qraw) {
  int wave = threadIdx.x >> 5, lane = threadIdx.x & 31;
  int tile = blockIdx.x * 4 + wave;            // 2048 tiles = 64 x 32
  int mt = tile >> 5, nt = tile & 31;
  int m = lane & 15, koff = (lane >> 4) << 1;
  const float* arow = x  + (mt * 16 + m) * C2; // A: rows of x
  const float* brow = Wq + (nt * 16 + m) * C2; // B[k][n] = Wq[n][k] -> row reads
  v8f c = {};
  for (int k0 = 0; k0 < C2; k0 += 4) {
    v2f a = *(const v2f*)(arow + k0 + koff);
    v2f b = *(const v2f*)(brow + k0 + koff);
    c = wmma4(a, b, c);
  }
  int col = nt * 16 + (lane & 15);
  float bias = bq[col];
  int rbase = mt * 16 + ((lane >> 4) ? 8 : 0);
  #pragma unroll
  for (int r = 0; r < 8; ++r)
    qraw[(rbase + r) * 512 + col] = c[r] + bias;
}

// ---------------- BatchNorm stats over 4096 rows, per channel ----------------
__global__ void k_bnstats(const float* __restrict__ qraw, float* __restrict__ meanp,
                          float* __restrict__ istdp) {
  __shared__ float s1[256], s2[256];
  int c = blockIdx.x, t = threadIdx.x;
  float a = 0.f, b = 0.f;
  for (int r = t; r < QR; r += 256) { float v = qraw[r * C2 + c]; a += v; b += v * v; }
  s1[t] = a; s2[t] = b; __syncthreads();
  for (int s = 128; s > 0; s >>= 1) {
    if (t < s) { s1[t] += s1[t + s]; s2[t] += s2[t + s]; }
    __syncthreads();
  }
  if (t == 0) {
    float mean = s1[0] / (float)QR;
    float var  = s2[0] / (float)QR - mean * mean;   // biased, like torch BN training
    meanp[c] = mean; istdp[c] = rsqrtf(var + EPSF);
  }
}

__global__ void k_bnapply(const float* __restrict__ qraw, const float* __restrict__ meanp,
                          const float* __restrict__ istdp, const float* __restrict__ g,
                          const float* __restrict__ be, float* __restrict__ qn,
                          float* __restrict__ outq) {
  int i = blockIdx.x * 256 + threadIdx.x;       // 524288 exact
  int c = i & (C2 - 1);
  float v = (qraw[i] - meanp[c]) * istdp[c] * g[c] + be[c];
  qn[i] = v; outq[i] = v;
}

// ---------------- v = emb(8192x128) @ Wv^T(128x128) + bv ----------------
__global__ void k_vproj(const float* __restrict__ emb, const float* __restrict__ Wv,
                        const float* __restrict__ bv, float* __restrict__ v) {
  int wave = threadIdx.x >> 5, lane = threadIdx.x & 31;
  int tile = blockIdx.x * 4 + wave;             // 4096 tiles = 512 x 8
  int mt = tile >> 3, nt = tile & 7;
  int m = lane & 15, koff = (lane >> 4) << 1;
  const float* arow = emb + (mt * 16 + m) * C2;
  const float* brow = Wv  + (nt * 16 + m) * C2;
  v8f c = {};
  for (int k0 = 0; k0 < C2; k0 += 4) {
    v2f a = *(const v2f*)(arow + k0 + koff);
    v2f b = *(const v2f*)(brow + k0 + koff);
    c = wmma4(a, b, c);
  }
  int col = nt * 16 + (lane & 15);
  float bias = bv[col];
  int rbase = mt * 16 + ((lane >> 4) ? 8 : 0);
  #pragma unroll
  for (int r = 0; r < 8; ++r)
    v[(rbase + r) * C2 + col] = c[r] + bias;
}

// ---------------- k = LayerNorm(emb) per row ----------------
__global__ void k_ln(const float* __restrict__ emb, const float* __restrict__ g,
                     const float* __restrict__ b, float* __restrict__ kout) {
  __shared__ float s1[128], s2[128];
  int row = blockIdx.x, t = threadIdx.x;
  float e = emb[row * C2 + t];
  s1[t] = e; s2[t] = e * e; __syncthreads();
  for (int s = 64; s > 0; s >>= 1) {
    if (t < s) { s1[t] += s1[t + s]; s2[t] += s2[t + s]; }
    __syncthreads();
  }
  float mean = s1[0] / (float)C2;
  float var  = s2[0] / (float)C2 - mean * mean;
  float istd = rsqrtf(var + EPSF);
  kout[row * C2 + t] = (e - mean) * istd * g[t] + b[t];
}

// -------- attn logits: qn(4096x128) @ k2(128x8192), k2[c][n]=kflat[c*8192+n] --------
__global__ void k_attn_logits(const float* __restrict__ qn, const float* __restrict__ kf,
                              float* __restrict__ attn) {
  int wave = threadIdx.x >> 5, lane = threadIdx.x & 31;
  int rt = blockIdx.y;                          // 256 row tiles
  int cb = blockIdx.x * 64 + wave * 16;         // 128 col groups of 64
  int m = lane & 15, koff = (lane >> 4) << 1;
  const float* arow = qn + (rt * 16 + m) * C2;
  const float* bcol = kf + cb + (lane & 15);    // column n, stride NN along k
  v8f c = {};
  for (int k0 = 0; k0 < C2; k0 += 4) {
    v2f a = *(const v2f*)(arow + k0 + koff);
    v2f b;
    b.x = bcol[(size_t)(k0 + koff) * NN];
    b.y = bcol[(size_t)(k0 + koff + 1) * NN];
    c = wmma4(a, b, c);
  }
  const float scale = 0.08838834764831845f;     // 1/sqrt(128)
  int col = cb + (lane & 15);
  int rbase = rt * 16 + ((lane >> 4) ? 8 : 0);
  #pragma unroll
  for (int r = 0; r < 8; ++r)
    attn[(size_t)(rbase + r) * NN + col] = c[r] * scale;
}

// ------- per-row softmax stats only: rowmax, rowinv = 1/sum(exp(x-max)) -------
__global__ void k_rowstats(const float* __restrict__ attn, float* __restrict__ rowmax,
                           float* __restrict__ rowinv) {
  __shared__ float red[256];
  size_t row = blockIdx.x; int t = threadIdx.x;
  const float* p = attn + row * NN;
  float mx = -3.0e38f;
  for (int i = t; i < NN; i += 256) mx = fmaxf(mx, p[i]);
  red[t] = mx; __syncthreads();
  for (int s = 128; s > 0; s >>= 1) { if (t < s) red[t] = fmaxf(red[t], red[t + s]); __syncthreads(); }
  mx = red[0]; __syncthreads();
  float sum = 0.f;
  for (int i = t; i < NN; i += 256) sum += __expf(p[i] - mx);   // row is L0/L2-hot
  red[t] = sum; __syncthreads();
  for (int s = 128; s > 0; s >>= 1) { if (t < s) red[t] += red[t + s]; __syncthreads(); }
  if (t == 0) { rowmax[row] = mx; rowinv[row] = 1.0f / red[0]; }
}

// --- applied = softmax(attn) @ v, consuming raw logits + (max, 1/sum) ---
// Async double-buffered LDS staging of the 16xCH logit chunk (ASYNCcnt pipeline).
__global__ void k_applied(const float* __restrict__ attn, const float* __restrict__ rowmax,
                          const float* __restrict__ rowinv, const float* __restrict__ v,
                          float* __restrict__ out2, float* __restrict__ out3) {
  __shared__ float raw[2][16 * CH];             // 2 x 8 KB logit chunks
  __shared__ float rinv[16];
  int wave = threadIdx.x >> 5, lane = threadIdx.x & 31;   // 8 waves
  int rt = blockIdx.x;                                    // 256 row tiles
  int nb = wave * 16;
  int m = lane & 15, koff = (lane >> 4) << 1;
  float mrow = rowmax[rt * 16 + m];             // A-operand row for this lane
  if (threadIdx.x < 16) rinv[threadIdx.x] = rowinv[rt * 16 + threadIdx.x];

  // staging assignment: thread t owns 8 consecutive floats of the 16xCH chunk
  int e  = threadIdx.x * 8;
  int sr = e >> 7;                               // chunk-local row 0..15
  int sx = e & (CH - 1);
  const float* gbase = attn + (size_t)(rt * 16 + sr) * NN + sx;
  unsigned int l0 = (unsigned int)(uintptr_t)&raw[0][e];
  unsigned int l1 = (unsigned int)(uintptr_t)&raw[1][e];

  // prologue: chunk 0 into buffer 0
  async_b128(l0, gbase);
  async_b128(l0 + 16, gbase + 4);

  const float* bcol = v + nb + (lane & 15);
  v8f acc = {};
  for (int c = 0; c < NCHUNK; ++c) {
    int buf = c & 1;
    if (c + 1 < NCHUNK) {                        // issue next chunk into other buffer
      const float* gn = gbase + (size_t)(c + 1) * CH;
      unsigned int ln = (buf ? l0 : l1);
      async_b128(ln, gn);
      async_b128(ln + 16, gn + 4);
      asm volatile("s_wait_asynccnt 0x2" ::: "memory");   // chunk c landed
    } else {
      asm volatile("s_wait_asynccnt 0x0" ::: "memory");
    }
    __syncthreads();
    const float* ar = &raw[buf][m * CH];
    int kg = c * CH;                             // global k base of this chunk
    for (int k0 = 0; k0 < CH; k0 += 4) {
      v2f a = *(const v2f*)(ar + k0 + koff);     // ds_load_b64
      v2f pa;
      pa.x = __expf(a.x - mrow);
      pa.y = __expf(a.y - mrow);
      v2f b;
      b.x = bcol[(kg + k0 + koff) * C2];         // V column, L2-resident
      b.y = bcol[(kg + k0 + koff + 1) * C2];
      acc = wmma4(pa, b, acc);
    }
    __syncthreads();                             // buffer reusable next iteration
  }
  int col = nb + (lane & 15);
  int half = (lane >> 4) ? 8 : 0;
  int rbase = rt * 16 + half;
  #pragma unroll
  for (int r = 0; r < 8; ++r) {
    float val = acc[r] * rinv[r + half];         // divide by softmax denominator
    out2[(rbase + r) * C2 + col] = val;
    out3[(rbase + r) * C2 + col] = val;
  }
}

// ------ top-40 per row, probabilities built on the fly from logits + stats ------
__global__ void k_topk(const float* __restrict__ attn, const float* __restrict__ rowmax,
                       const float* __restrict__ rowinv, float* __restrict__ topv,
                       int* __restrict__ topi, int* __restrict__ rowsum) {
  __shared__ float vals[NN];
  __shared__ float rv[256];
  __shared__ int   ri[256];
  __shared__ int   chosen[TK];
  size_t row = blockIdx.x; int t = threadIdx.x;
  const float* p = attn + row * NN;
  float mx = rowmax[row], inv = rowinv[row];
  for (int i = t; i < NN; i += 256) vals[i] = __expf(p[i] - mx) * inv;
  __syncthreads();
  for (int j = 0; j < TK; ++j) {
    float bm = -1.0f; int bi = 0;
    for (int i = t; i < NN; i += 256) { float v = vals[i]; if (v > bm) { bm = v; bi = i; } }
    rv[t] = bm; ri[t] = bi; __syncthreads();
    for (int s = 128; s > 0; s >>= 1) {
      if (t < s && rv[t + s] > rv[t]) { rv[t] = rv[t + s]; ri[t] = ri[t + s]; }
      __syncthreads();
    }
    if (t == 0) {
      int idx = ri[0];
      topv[row * TK + j] = rv[0];
      topi[row * TK + j] = idx;
      chosen[j] = idx;
      vals[idx] = -1.0f;
    }
    __syncthreads();
  }
  if (t == 0) {
    int s = 0;
    for (int j = 0; j < TK; ++j) s += chosen[j];
    rowsum[row] = s;
  }
}

// ---------------- per-batch permutation seed ----------------
__global__ void k_seed(const int* __restrict__ rowsum, int* __restrict__ seed) {
  int b = threadIdx.x;                         // 256 threads, 1 block
  long long s = 0;
  for (int r = 0; r < MM; ++r) s += rowsum[b * MM + r];
  float avg = (float)s / (float)(MM * TK);
  seed[b] = ((int)floorf(avg / (NN / 4.0f))) % 4;
}

// ---------------- deterministic permutation tables + inverses ----------------
__global__ void k_perms(int* __restrict__ perm, int* __restrict__ iperm) {
  if (threadIdx.x != 0) return;
  int s = blockIdx.x;                          // 4 tables
  int* p = perm + s * NN;
  for (int i = 0; i < NN; ++i) p[i] = i;
  unsigned long long st = 0x9E3779B97F4A7C15ull * (unsigned long long)(s + 1)
                        + 0x243F6A8885A308D3ull;
  for (int i = NN - 1; i > 0; --i) {
    st ^= st << 13; st ^= st >> 7; st ^= st << 17;
    int j = (int)(st % (unsigned long long)(i + 1));
    int tmp = p[i]; p[i] = p[j]; p[j] = tmp;
  }
  int* ip = iperm + s * NN;
  for (int i = 0; i < NN; ++i) ip[p[i]] = i;
}

__global__ void k_zero(float* __restrict__ out, size_t n) {
  size_t i = (size_t)blockIdx.x * blockDim.x + threadIdx.x;
  size_t stride = (size_t)gridDim.x * blockDim.x;
  for (; i < n; i += stride) out[i] = 0.0f;
}

// attn_w[b,q,n] = masked[b,q,perm[n]] -> scatter topv at n = invperm[topi]
__global__ void k_scatter(const float* __restrict__ topv, const int* __restrict__ topi,
                          const int* __restrict__ seed, const int* __restrict__ iperm,
                          float* __restrict__ out0) {
  int t = blockIdx.x * 256 + threadIdx.x;      // 163840 exact
  int row = t / TK, j = t - row * TK;
  int b = row >> 4;
  int i = topi[row * TK + j];
  int n = iperm[seed[b] * NN + i];
  out0[(size_t)row * NN + n] = topv[row * TK + j];
}

extern "C" void kernel_launch(void* const* d_in, const int* in_sizes, int n_in,
                              void* d_out, int out_size, void* d_ws, size_t ws_size,
                              hipStream_t stream) {
  const float* x    = (const float*)d_in[0];
  const float* emb  = (const float*)d_in[1];
  const float* Wq   = (const float*)d_in[2];
  const float* bq   = (const float*)d_in[3];
  const float* Wv   = (const float*)d_in[4];
  const float* bv   = (const float*)d_in[5];
  const float* bn_g = (const float*)d_in[6];
  const float* bn_b = (const float*)d_in[7];
  const float* ln_g = (const float*)d_in[8];
  const float* ln_b = (const float*)d_in[9];

  float* ws = (float*)d_ws;
  float* qraw  = ws;                    // [4096,128]
  float* qn    = ws + 524288;           // [4096,128]
  float* v     = ws + 1048576;          // [8192,128]
  float* kf    = ws + 2097152;          // [8192*128] flat, viewed as k2[128,8192]
  float* meanp = ws + 3145728;          // [128]
  float* istdp = ws + 3145856;          // [128]
  float* rowmax= ws + 3145984;          // [4096]
  float* rowinv= ws + 3150080;          // [4096]
  float* topv  = ws + 3154176;          // [4096,40]
  int*   topi  = (int*)(ws + 3318016);  // [4096,40]
  int*   rowsum= (int*)(ws + 3481856);  // [4096]
  int*   seed  = (int*)(ws + 3485952);  // [256]
  int*   perm  = (int*)(ws + 3486208);  // [4,8192]
  int*   iperm = (int*)(ws + 3518976);  // [4,8192]
  float* attn  = ws + 3551744;          // [4096,8192] raw logits

  float* out0 = (float*)d_out;          // attn_w_flat [256, 131072]
  float* out1 = out0 + 33554432;        // q           [4096,128]
  float* out2 = out1 + 524288;          // applied     [4096,128]
  float* out3 = out2 + 524288;          // applied_flat

  k_qraw<<<512, 128, 0, stream>>>(x, Wq, bq, qraw);
  k_bnstats<<<128, 256, 0, stream>>>(qraw, meanp, istdp);
  k_bnapply<<<2048, 256, 0, stream>>>(qraw, meanp, istdp, bn_g, bn_b, qn, out1);
  k_vproj<<<1024, 128, 0, stream>>>(emb, Wv, bv, v);
  k_ln<<<8192, 128, 0, stream>>>(emb, ln_g, ln_b, kf);
  k_attn_logits<<<dim3(128, 256), 128, 0, stream>>>(qn, kf, attn);
  k_rowstats<<<4096, 256, 0, stream>>>(attn, rowmax, rowinv);
  k_applied<<<256, 256, 0, stream>>>(attn, rowmax, rowinv, v, out2, out3);
  k_topk<<<4096, 256, 0, stream>>>(attn, rowmax, rowinv, topv, topi, rowsum);
  k_seed<<<1, 256, 0, stream>>>(rowsum, seed);
  k_perms<<<4, 32, 0, stream>>>(perm, iperm);
  k_zero<<<2048, 256, 0, stream>>>(out0, (size_t)33554432);
  k_scatter<<<640, 256, 0, stream>>>(topv, topi, seed, iperm, out0);
}